// PCENLayer_28183575396437
// MI455X (gfx1250) — compile-verified
//
#include <hip/hip_runtime.h>

// PCEN: B=64, T=4096, C=80, f32 in/out.
// EMA scan done as chained V_WMMA_F32_16X16X4_F32 (prefix-sum matmul per 16-step tile).

#define BB 64
#define TT 4096
#define CC 80
#define NCHUNK 32
#define LCH 128              // chunk length (T = NCHUNK * LCH)
#define NTILE (LCH / 16)     // 8 WMMA tiles per chunk
#define NCT 5                // channel tiles of 16 (C = 80)
#define DMIN 0.00390625f     // clamp decay so d^-15 <= 2^120 stays finite in f32
#define FLOORV 1e-6f

typedef __attribute__((ext_vector_type(2))) float v2f;
typedef __attribute__((ext_vector_type(8))) float v8f;

__device__ __forceinline__ float clamp01f(float x) { return fminf(fmaxf(x, 0.0f), 1.0f); }
// Raw CDNA5 transcendentals: v_log_f32 is log2, v_exp_f32 is exp2.
__device__ __forceinline__ float flog2(float x) { return __builtin_amdgcn_logf(x); }
__device__ __forceinline__ float fexp2(float x) { return __builtin_amdgcn_exp2f(x); }

// -------- Kernel 1: per-chunk zero-state scan responses (chunk aggregates) --------
__global__ __launch_bounds__(256)
void pcen_chunk_agg(const float* __restrict__ X, const float* __restrict__ smooth,
                    float* __restrict__ bchunk) {
  const int lane  = threadIdx.x & 31;
  const int wid   = blockIdx.x * 8 + (threadIdx.x >> 5);
  const int ctile = wid % NCT;
  const int tmp   = wid / NCT;
  const int chunk = tmp % NCHUNK;
  const int b     = tmp / NCHUNK;
  const int hi    = lane >> 4;            // 0: lanes 0-15, 1: lanes 16-31
  const int c     = ctile * 16 + (lane & 15);

  const float wsm = clamp01f(smooth[c]);
  const float d   = fminf(fmaxf(1.0f - wsm, DMIN), 1.0f);
  const float l2d = flog2(d);

  // Per-lane constants. A/B K index for vgpr j (of 8): u = 4*(j>>1) + (j&1) + 2*hi.
  float aP[8], syB[8];
#pragma unroll
  for (int j = 0; j < 8; ++j) {
    const int u = 4 * (j >> 1) + (j & 1) + 2 * hi;
    aP[j]  = (u <= (lane & 15)) ? 1.0f : 0.0f;       // ones lower-triangular (prefix)
    syB[j] = wsm * fexp2(-(float)u * l2d);           // w * d^-u
  }
  const float dt7 = fexp2((float)(7 + 8 * hi) * l2d);  // d^t, t = 7 (lo) / 15 (hi)

  // Per-lane base pointer for B-layout loads (row t0 + u, channel c); u's hi part folded in.
  const float* pB = X + ((size_t)b * TT + (size_t)chunk * LCH + 2 * hi) * CC + c;

  float s = 0.0f;  // zero-state response
#pragma unroll 1
  for (int tile = 0; tile < NTILE; ++tile) {
    float xb[8];
#pragma unroll
    for (int j = 0; j < 8; ++j) {
      const int u0 = 4 * (j >> 1) + (j & 1);         // 0,1,4,5,8,9,12,13
      xb[j] = pB[(size_t)u0 * CC];
    }
    v8f acc = {0.f, 0.f, 0.f, 0.f, 0.f, 0.f, 0.f, 0.f};
#pragma unroll
    for (int kk = 0; kk < 4; ++kk) {
      v2f a2; a2[0] = aP[2 * kk];               a2[1] = aP[2 * kk + 1];
      v2f b2; b2[0] = xb[2 * kk] * syB[2 * kk]; b2[1] = xb[2 * kk + 1] * syB[2 * kk + 1];
      acc = __builtin_amdgcn_wmma_f32_16x16x4_f32(false, a2, false, b2, (short)0, acc,
                                                  false, false);
    }
    // a[15] lives in D vgpr 7 on the hi lanes; broadcast to both halves per channel.
    const float a15 = dt7 * (acc[7] + d * s);
    s = __shfl(a15, (lane & 15) + 16, 32);
    pB += 16 * CC;
  }
  if (lane < 16) bchunk[((size_t)b * NCHUNK + chunk) * CC + c] = s;
}

// -------- Kernel 2: sequential combine over chunks -> per-chunk start states --------
__global__ __launch_bounds__(256)
void pcen_combine(const float* __restrict__ X, const float* __restrict__ smooth,
                  const float* __restrict__ bchunk, float* __restrict__ start) {
  const int tid = blockIdx.x * 256 + threadIdx.x;
  if (tid >= BB * CC) return;
  const int b = tid / CC, c = tid % CC;
  const float wsm = clamp01f(smooth[c]);
  const float d   = fminf(fmaxf(1.0f - wsm, DMIN), 1.0f);
  const float dL  = fexp2((float)LCH * flog2(d));
  // Fold init a0 = x0 as virtual start state s_-1 = x0  (w*x0 + d*x0 = x0).
  float s = X[(size_t)b * TT * CC + c];
  for (int k = 0; k < NCHUNK; ++k) {
    start[((size_t)b * NCHUNK + k) * CC + c] = s;
    s = dL * s + bchunk[((size_t)b * NCHUNK + k) * CC + c];
  }
}

// -------- Kernel 3: re-run tiles with true start states, apply PCEN, store --------
__global__ __launch_bounds__(256)
void pcen_emit(const float* __restrict__ X, const float* __restrict__ alpha,
               const float* __restrict__ delta, const float* __restrict__ root,
               const float* __restrict__ smooth, const float* __restrict__ start,
               float* __restrict__ out) {
  const int lane  = threadIdx.x & 31;
  const int wid   = blockIdx.x * 8 + (threadIdx.x >> 5);
  const int ctile = wid % NCT;
  const int tmp   = wid / NCT;
  const int chunk = tmp % NCHUNK;
  const int b     = tmp / NCHUNK;
  const int hi    = lane >> 4;
  const int c     = ctile * 16 + (lane & 15);

  const float wsm = clamp01f(smooth[c]);
  const float d   = fminf(fmaxf(1.0f - wsm, DMIN), 1.0f);
  const float l2d = flog2(d);
  const float ac  = fminf(alpha[c], 1.0f);
  const float dc  = delta[c];
  const float ir  = 1.0f / fmaxf(root[c], 1.0f);
  const float off = fexp2(ir * flog2(dc));  // delta^(1/r)

  float aP[8], syB[8], dt[8];
#pragma unroll
  for (int j = 0; j < 8; ++j) {
    const int u = 4 * (j >> 1) + (j & 1) + 2 * hi;
    aP[j]  = (u <= (lane & 15)) ? 1.0f : 0.0f;
    syB[j] = wsm * fexp2(-(float)u * l2d);
    dt[j]  = fexp2((float)(j + 8 * hi) * l2d);  // d^t for D row t = j + 8*hi
  }

  float s = start[((size_t)b * NCHUNK + chunk) * CC + c];

  const size_t rowBase = ((size_t)b * TT + (size_t)chunk * LCH) * CC + c;
  const float* pB = X + rowBase + (size_t)(2 * hi) * CC;  // B-operand layout
  const float* pD = X + rowBase + (size_t)(8 * hi) * CC;  // D/C-operand layout
  float*       pO = out + rowBase + (size_t)(8 * hi) * CC;

#pragma unroll 1
  for (int tile = 0; tile < NTILE; ++tile) {
    float xb[8];
#pragma unroll
    for (int j = 0; j < 8; ++j) {
      const int u0 = 4 * (j >> 1) + (j & 1);
      xb[j] = pB[(size_t)u0 * CC];
    }
    v8f acc = {0.f, 0.f, 0.f, 0.f, 0.f, 0.f, 0.f, 0.f};
#pragma unroll
    for (int kk = 0; kk < 4; ++kk) {
      v2f a2; a2[0] = aP[2 * kk];               a2[1] = aP[2 * kk + 1];
      v2f b2; b2[0] = xb[2 * kk] * syB[2 * kk]; b2[1] = xb[2 * kk + 1] * syB[2 * kk + 1];
      acc = __builtin_amdgcn_wmma_f32_16x16x4_f32(false, a2, false, b2, (short)0, acc,
                                                  false, false);
    }
    float a7 = 0.0f;
#pragma unroll
    for (int v = 0; v < 8; ++v) {
      const float ema = dt[v] * (acc[v] + d * s);  // true EMA at (t0+v+8*hi, c)
      if (v == 7) a7 = ema;
      const float xv  = pD[(size_t)v * CC];
      const float lp  = flog2(ema + FLOORV);
      const float num = xv * fexp2(-ac * lp) + dc;     // x/(eps+ema)^a + delta
      pO[(size_t)v * CC] = fexp2(ir * flog2(num)) - off;
    }
    s = __shfl(a7, (lane & 15) + 16, 32);  // carry a[15] per channel
    pB += 16 * CC; pD += 16 * CC; pO += 16 * CC;
  }
}

extern "C" void kernel_launch(void* const* d_in, const int* in_sizes, int n_in,
                              void* d_out, int out_size, void* d_ws, size_t ws_size,
                              hipStream_t stream) {
  const float* X      = (const float*)d_in[0];
  const float* alpha  = (const float*)d_in[1];
  const float* delta  = (const float*)d_in[2];
  const float* root   = (const float*)d_in[3];
  const float* smooth = (const float*)d_in[4];
  float* out = (float*)d_out;

  float* bchunk = (float*)d_ws;                       // [B][NCHUNK][C]
  float* start  = bchunk + (size_t)BB * NCHUNK * CC;  // [B][NCHUNK][C]

  const int nwaves = BB * NCHUNK * NCT;  // 10240
  pcen_chunk_agg<<<nwaves / 8, 256, 0, stream>>>(X, smooth, bchunk);
  pcen_combine<<<(BB * CC + 255) / 256, 256, 0, stream>>>(X, smooth, bchunk, start);
  pcen_emit<<<nwaves / 8, 256, 0, stream>>>(X, alpha, delta, root, smooth, start, out);
}